// Expander_44470091383230
// MI455X (gfx1250) — compile-verified
//
#include <hip/hip_runtime.h>

typedef __attribute__((ext_vector_type(2))) float v2f;
typedef __attribute__((ext_vector_type(8))) float v8f;
typedef __attribute__((ext_vector_type(4))) float f4;

#define NQ  375   // real class columns: 3 out-ch * 5*5*5
#define NQP 384   // padded to 24 tiles of 16 columns

// ---------------------------------------------------------------------------
// Kernel A: build effective affine table Eb[NQP][4] (cols 0..2 = per-input-
// channel weight, col 3 = bias) by simulating conv1->conv2 on a canonical
// 6x6x6 volume. Class reps {0,1,2,4,5} map exactly to the 5 boundary classes
// of any dim >= 6 (here 16 and 28).
// ---------------------------------------------------------------------------
__global__ __launch_bounds__(256) void build_eff(
    const float* __restrict__ w1, const float* __restrict__ b1,
    const float* __restrict__ w2, const float* __restrict__ b2,
    float* __restrict__ Eb) {
  // t1[field(4)][o(4)][z(6)][y(6)][x(6)] ; field 0..2 = basis ci, 3 = bias
  __shared__ float t1[4 * 4 * 216];
  const int tid = threadIdx.x;

  // Stage 1: conv1 applied to basis/bias fields (3456 entries)
  for (int i = tid; i < 3456; i += 256) {
    int xx = i % 6, yy = (i / 6) % 6, zz = (i / 36) % 6;
    int o = (i / 216) % 4, f = i / 864;
    float v;
    if (f == 3) {
      v = b1[o];  // conv1(0) + b1 is spatially constant
    } else {
      v = 0.0f;
      for (int kd = 0; kd < 3; ++kd) {
        int z = zz + kd - 1; if (z < 0 || z > 5) continue;
        for (int kh = 0; kh < 3; ++kh) {
          int y = yy + kh - 1; if (y < 0 || y > 5) continue;
          for (int kw = 0; kw < 3; ++kw) {
            int x = xx + kw - 1; if (x < 0 || x > 5) continue;
            v += w1[(o * 3 + f) * 27 + kd * 9 + kh * 3 + kw];
          }
        }
      }
    }
    t1[i] = v;
  }
  __syncthreads();

  // Stage 2: conv2 sampled at the 125 class-representative positions
  const int rep[5] = {0, 1, 2, 4, 5};
  for (int i = tid; i < NQP * 4; i += 256) {
    int q = i >> 2, k = i & 3;
    float v = 0.0f;
    if (q < NQ) {
      int wc = q % 5, hc = (q / 5) % 5, dc = (q / 25) % 5, co = q / 125;
      int zi = rep[dc], yi = rep[hc], xi = rep[wc];
      for (int o = 0; o < 4; ++o) {
        for (int kd = 0; kd < 3; ++kd) {
          int z = zi + kd - 1; if (z < 0 || z > 5) continue;
          for (int kh = 0; kh < 3; ++kh) {
            int y = yi + kh - 1; if (y < 0 || y > 5) continue;
            for (int kw = 0; kw < 3; ++kw) {
              int x = xi + kw - 1; if (x < 0 || x > 5) continue;
              v += w2[(co * 4 + o) * 27 + kd * 9 + kh * 3 + kw] *
                   t1[k * 864 + o * 216 + z * 36 + y * 6 + x];
            }
          }
        }
      }
      if (k == 3) v += b2[co];
    }
    Eb[i] = v;  // Eb[q][k], zero-padded for q >= NQ
  }
}

// ---------------------------------------------------------------------------
// Kernel B: Y[1024][NQP] = Xa[1024][4] @ Eb[4][NQP] via V_WMMA_F32_16X16X4_F32
// Homogeneous trick: Xa[:,3] = 1.0, Eb row 3 = bias K[q].
// One wave per 16x16 tile; 64 row-tiles x 24 col-tiles = 1536 waves.
// A layout (32-bit 16x4): lane%16 -> M, lane/16 selects K-pair {0,1}/{2,3}.
// ---------------------------------------------------------------------------
__global__ __launch_bounds__(128) void gemm_wmma(
    const float* __restrict__ x, const float* __restrict__ Eb,
    float* __restrict__ Y) {
  int wave  = (blockIdx.x * 128 + threadIdx.x) >> 5;  // 0..1535
  int lane  = threadIdx.x & 31;
  int btile = wave / 24;
  int ctile = wave % 24;
  int m     = lane & 15;
  int khalf = lane >> 4;
  int k0    = khalf * 2;            // 0 or 2

  int row = btile * 16 + m;         // batch row 0..1023
  v2f a, b;
  a.x = x[row * 3 + k0];                                   // k0 in {0,2}
  a.y = (k0 == 2) ? 1.0f : x[row * 3 + k0 + 1];            // k=3 -> 1.0

  int q = ctile * 16 + m;           // padded column
  b.x = Eb[q * 4 + k0];
  b.y = Eb[q * 4 + k0 + 1];

  v8f c = {};
  // (neg_a, A, neg_b, B, c_mod, C, reuse_a, reuse_b)
  c = __builtin_amdgcn_wmma_f32_16x16x4_f32(
      false, a, false, b, (short)0, c, false, false);

  // D layout: lane%16 -> N, VGPR r -> M = r + 8*(lane/16)
  #pragma unroll
  for (int r = 0; r < 8; ++r) {
    int mm = r + 8 * khalf;
    Y[(btile * 16 + mm) * NQP + ctile * 16 + m] = c[r];
  }
}

// ---------------------------------------------------------------------------
// Kernel C: broadcast-expand Y classes to the full (1024,3,16,28,28) output.
// Pure store-bandwidth: one float4 (GLOBAL_STORE_B128) per thread, coalesced,
// non-temporal (154 MB streamed once, nothing re-reads it in the timed path).
// ---------------------------------------------------------------------------
__device__ __forceinline__ int cls5(int i, int n) {
  return i == 0 ? 0 : (i == 1 ? 1 : (i == n - 1 ? 4 : (i == n - 2 ? 3 : 2)));
}

__global__ __launch_bounds__(256) void expand(
    const float* __restrict__ Y, float* __restrict__ out) {
  int tid = blockIdx.x * 256 + threadIdx.x;  // 0 .. 9,633,791 (fits int32)
  int wg = tid % 7;
  int t  = tid / 7;
  int h  = t % 28; t /= 28;
  int d  = t % 16; t /= 16;
  int co = t % 3;
  int b  = t / 3;

  int dc = cls5(d, 16);
  int hc = cls5(h, 28);
  const float* Yr = Y + b * NQP + (co * 5 + dc) * 25 + hc * 5;

  int w0 = wg * 4;
  f4 v;
  v.x = Yr[cls5(w0 + 0, 28)];
  v.y = Yr[cls5(w0 + 1, 28)];
  v.z = Yr[cls5(w0 + 2, 28)];
  v.w = Yr[cls5(w0 + 3, 28)];
  __builtin_nontemporal_store(v, (f4*)(out + (size_t)tid * 4));
}

// ---------------------------------------------------------------------------
extern "C" void kernel_launch(void* const* d_in, const int* in_sizes, int n_in,
                              void* d_out, int out_size, void* d_ws, size_t ws_size,
                              hipStream_t stream) {
  const float* x  = (const float*)d_in[0];  // (1024,3)
  const float* w1 = (const float*)d_in[1];  // (4,3,3,3,3)
  const float* b1 = (const float*)d_in[2];  // (4,)
  const float* w2 = (const float*)d_in[3];  // (3,4,3,3,3)
  const float* b2 = (const float*)d_in[4];  // (3,)
  float* out = (float*)d_out;               // 1024*3*16*28*28

  float* Eb = (float*)d_ws;                 // NQP*4 floats        (6 KB)
  float* Y  = Eb + NQP * 4;                 // 1024*NQP floats     (1.5 MB)

  build_eff<<<1, 256, 0, stream>>>(w1, b1, w2, b2, Eb);
  gemm_wmma<<<(1536 * 32) / 128, 128, 0, stream>>>(x, Eb, Y);   // 384 blocks
  expand<<<(1024 * 3 * 16 * 28 * 28) / (4 * 256), 256, 0, stream>>>(Y, out); // 37632 blocks
}